// LTVMinimumPhaseFilter_46523085750323
// MI455X (gfx1250) — compile-verified
//
#include <hip/hip_runtime.h>

typedef float v2f __attribute__((ext_vector_type(2)));
typedef float v8f __attribute__((ext_vector_type(8)));

#define B_    16
#define T_    256000
#define HOP   128
#define W_    512
#define ORD   16
#define NF    2000            // frames per batch (== n_frames)
#define PAD   192             // (W - HOP)/2
#define TP    (T_ + 2*PAD)    // padded length per batch = 256384
#define NBLK  32              // 512 / 16 blocks per frame

__global__ __launch_bounds__(256) void zero_acc_kernel(float* __restrict__ acc, int n) {
  int i = blockIdx.x * blockDim.x + threadIdx.x;
  if (i < n) acc[i] = 0.0f;
}

// One wave32 per frame. Blocked state-space LPC synthesis:
//   y_blk = H*(g*x_blk) - G*p ,  H = truncated impulse response (16x16 lower-tri),
//   G = H*C with C[i][j] = a[i+j] (i+j<16).  H*C and H*GX done with V_WMMA_F32_16X16X4_F32.
__global__ __launch_bounds__(32)
void lpc_frame_kernel(const float* __restrict__ ex,
                      const float* __restrict__ gain,
                      const float* __restrict__ acoef,
                      const float* __restrict__ window,
                      float* __restrict__ acc,
                      int residue)
{
  __shared__ float al[16];          // LPC coefficients (dyn-indexable)
  __shared__ float hl[16];          // impulse response (dyn-indexable)
  __shared__ float gx[W_];          // g*x, linear index s = n*16 + k  (n=block, k=row)
  __shared__ float ypl[16 * NBLK];  // Yp[i][n] = (H*GX)[i][n]
  __shared__ float g16[16 * 16];    // G[i][j]
  __shared__ float yl[W_];          // filtered frame output

  const int tid  = threadIdx.x;
  const int gid  = blockIdx.x;            // 0 .. B*(NF/4)-1
  const int b    = gid / (NF / 4);
  const int fr   = residue + 4 * (gid % (NF / 4));
  const int fidx = b * NF + fr;

  if (tid < ORD) al[tid] = acoef[fidx * ORD + tid];
  const float g = gain[fidx];
  __syncthreads();

  // impulse response of 1/(1 + sum a_k z^-k), computed redundantly per lane (constant idx)
  float h[16];
  h[0] = 1.0f;
#pragma unroll
  for (int n = 1; n < 16; ++n) {
    float s = 0.0f;
#pragma unroll
    for (int k = 1; k <= n; ++k) s += al[k - 1] * h[n - k];
    h[n] = -s;
  }
#pragma unroll
  for (int n = 0; n < 16; ++n)
    if (tid == n) hl[n] = h[n];

  // stage g*x (coalesced), honoring the reference's zero padding of ex
  const int base = fr * HOP - PAD;
#pragma unroll
  for (int i = 0; i < 16; ++i) {
    int s  = i * 32 + tid;
    int xi = base + s;
    float xv = 0.0f;
    if (xi >= 0 && xi < T_) xv = ex[b * T_ + xi];
    gx[s] = g * xv;
  }
  __syncthreads();

  const int half = tid >> 4;   // K-half for A/B fragments
  const int ln   = tid & 15;   // M (A) / N (B) index

  v8f gAcc = {0, 0, 0, 0, 0, 0, 0, 0};
  v8f y0   = {0, 0, 0, 0, 0, 0, 0, 0};
  v8f y1   = {0, 0, 0, 0, 0, 0, 0, 0};

#pragma unroll
  for (int kc = 0; kc < 4; ++kc) {
    const int j0 = kc * 4 + half * 2;   // K index of fragment element .x ; .y is j0+1

    // A fragment: H[M][K] = (M>=K) ? h[M-K] : 0
    v2f aF;
    {
      int i0 = ln - j0;
      int i1 = i0 - 1;
      float v0 = hl[i0 & 15];
      float v1 = hl[i1 & 15];
      aF.x = (i0 >= 0) ? v0 : 0.0f;
      aF.y = (i1 >= 0) ? v1 : 0.0f;
    }

    // B fragment for C: C[K][N] = (K+N<=15) ? a[K+N] : 0
    v2f cF;
    {
      int s0 = j0 + ln;
      int s1 = s0 + 1;
      float v0 = al[s0 & 15];
      float v1 = al[s1 & 15];
      cF.x = (s0 <= 15) ? v0 : 0.0f;
      cF.y = (s1 <= 15) ? v1 : 0.0f;
    }

    // B fragments for GX: GX[K][N] = gx[N*16 + K], two N-halves (blocks 0-15 / 16-31)
    v2f bF0, bF1;
    bF0.x = gx[ln * 16 + j0];
    bF0.y = gx[ln * 16 + j0 + 1];
    bF1.x = gx[(ln + 16) * 16 + j0];
    bF1.y = gx[(ln + 16) * 16 + j0 + 1];

    gAcc = __builtin_amdgcn_wmma_f32_16x16x4_f32(false, aF, false, cF, (short)0, gAcc, false, false);
    y0   = __builtin_amdgcn_wmma_f32_16x16x4_f32(false, aF, false, bF0, (short)0, y0, false, false);
    y1   = __builtin_amdgcn_wmma_f32_16x16x4_f32(false, aF, false, bF1, (short)0, y1, false, false);
  }

  // scatter D fragments to LDS: VGPR v holds row M = v + 8*half, col = ln
#pragma unroll
  for (int v = 0; v < 8; ++v) {
    int row = v + 8 * half;
    g16[row * 16 + ln]        = gAcc[v];
    ypl[row * NBLK + ln]      = y0[v];
    ypl[row * NBLK + 16 + ln] = y1[v];
  }
  __syncthreads();

  // each lane keeps one row of G in registers (lanes 16-31 mirror rows 0-15)
  float Grow[16];
#pragma unroll
  for (int j = 0; j < 16; ++j) Grow[j] = g16[ln * 16 + j];

  // short serial sweep over 32 blocks: y_b = Yp[:,b] - G * p ,  p[j] = y_{b-1}[15-j]
  for (int blk = 0; blk < NBLK; ++blk) {
    float v = ypl[ln * NBLK + blk];
    if (blk > 0) {
      const int pb = (blk - 1) * 16 + 15;
#pragma unroll
      for (int j = 0; j < 16; ++j) v -= Grow[j] * yl[pb - j];  // broadcast LDS reads
    }
    if (tid < 16) yl[blk * 16 + ln] = v;
    __syncthreads();
  }

  // window + overlap-add; residue partition (hop*4 == W) makes frames in this
  // launch disjoint -> plain read-modify-write, deterministic, no atomics
#pragma unroll
  for (int i = 0; i < 16; ++i) {
    int s = i * 32 + tid;
    float wv = yl[s] * window[s];
    int pos = b * TP + fr * HOP + s;
    acc[pos] += wv;
  }
}

__global__ __launch_bounds__(256)
void finalize_kernel(const float* __restrict__ acc,
                     const float* __restrict__ window,
                     float* __restrict__ out)
{
  int i = blockIdx.x * blockDim.x + threadIdx.x;
  if (i >= B_ * T_) return;
  int b = i / T_;
  int t = i - b * T_;
  int tp = t + PAD;

  int fhi = tp >> 7;            // tp / HOP
  if (fhi > NF - 1) fhi = NF - 1;
  int flo = fhi - 3;
  if (flo < 0) flo = 0;

  float norm = 0.0f;
  for (int f = flo; f <= fhi; ++f) {
    int d = tp - f * HOP;       // >= 0 by construction
    if (d < W_) norm += window[d];
  }
  out[i] = acc[b * TP + tp] / norm;
}

extern "C" void kernel_launch(void* const* d_in, const int* in_sizes, int n_in,
                              void* d_out, int out_size, void* d_ws, size_t ws_size,
                              hipStream_t stream) {
  (void)in_sizes; (void)n_in; (void)out_size; (void)ws_size;
  const float* ex     = (const float*)d_in[0];
  const float* gain   = (const float*)d_in[1];
  const float* acoef  = (const float*)d_in[2];
  const float* window = (const float*)d_in[3];
  // d_in[4] = hop_length scalar (hardcoded to 128)

  float* acc = (float*)d_ws;       // B*TP floats = ~16.4 MB
  float* out = (float*)d_out;

  const int nacc = B_ * TP;
  zero_acc_kernel<<<(nacc + 255) / 256, 256, 0, stream>>>(acc, nacc);

  for (int r = 0; r < 4; ++r) {
    lpc_frame_kernel<<<B_ * (NF / 4), 32, 0, stream>>>(ex, gain, acoef, window, acc, r);
  }

  const int nout = B_ * T_;
  finalize_kernel<<<(nout + 255) / 256, 256, 0, stream>>>(acc, window, out);
}